// Views2Points_46969762349792
// MI455X (gfx1250) — compile-verified
//
#include <hip/hip_runtime.h>

// Problem constants (match reference)
#define BATCH      32
#define NPTS       8192
#define NUM_GROUP  512
#define GROUP_SIZE 32

// KNN kernel tiling
#define CTILE        16                      // centers per workgroup (one WMMA M-tile)
#define WAVES_KNN    8                       // 256 threads
#define NT           4                       // 16-point tiles per wave per iteration
#define PTS_PER_ITER (WAVES_KNN * 16 * NT)   // 512 points per iteration
#define NITER        (NPTS / PTS_PER_ITER)   // 16
#define SSTRIDE      (PTS_PER_ITER + 4)      // 516: 8*516 % 64 == 32 -> conflict-free scatter

typedef float v2f __attribute__((ext_vector_type(2)));
typedef float v8f __attribute__((ext_vector_type(8)));

// ---------------------------------------------------------------------------
// wave32 helpers
// ---------------------------------------------------------------------------
__device__ __forceinline__ float bcast_f(float v, int src) {
  return __uint_as_float(__builtin_amdgcn_readlane(__float_as_uint(v), src));
}
__device__ __forceinline__ int bcast_i(int v, int src) {
  return __builtin_amdgcn_readlane(v, src);
}

__device__ __forceinline__ void wave_argmax(float& d, int& i) {
  // all-lanes max with lowest-index tie-break; result uniform across wave32
  #pragma unroll
  for (int off = 16; off > 0; off >>= 1) {
    float od = __shfl_xor(d, off);
    int   oi = __shfl_xor(i, off);
    if (od > d || (od == d && oi < i)) { d = od; i = oi; }
  }
}

// Distributed top-K (K = 32 = one candidate per lane).
// cd/ci: candidate per lane; thr: uniform current max of cd.
__device__ __forceinline__ void topk_insert(float dnew, int inew,
                                            float& cd, int& ci, float& thr,
                                            int lane) {
  unsigned mask = __builtin_amdgcn_ballot_w32(dnew < thr);
  while (mask) {
    int src = __builtin_ctz(mask);
    mask &= mask - 1;
    float dd = bcast_f(dnew, src);   // v_readlane: src is wave-uniform
    int   ii = bcast_i(inew, src);
    if (dd < thr) {
      // evict current max candidate
      float md_ = cd; int mi_ = lane;
      wave_argmax(md_, mi_);
      if (lane == mi_) { cd = dd; ci = ii; }
      // refresh threshold
      float td = cd; int ti = lane;
      wave_argmax(td, ti);
      thr = td;
    }
  }
}

// 32-lane bitonic sort ascending by (d, i) lexicographic.
__device__ __forceinline__ void bitonic32(float& d, int& i, int lane) {
  #pragma unroll
  for (int k = 2; k <= 32; k <<= 1) {
    #pragma unroll
    for (int j = k >> 1; j > 0; j >>= 1) {
      float od = __shfl_xor(d, j);
      int   oi = __shfl_xor(i, j);
      bool up           = ((lane & k) == 0);   // k==32: always ascending
      bool lower        = ((lane & j) == 0);
      bool otherSmaller = (od < d) || (od == d && oi < i);
      bool takeOther    = (lower == up) ? otherSmaller : !otherSmaller;
      if (takeOther) { d = od; i = oi; }
    }
  }
}

// ---------------------------------------------------------------------------
// Kernel 1: farthest point sampling. One block per batch; all points in LDS.
// Writes center coords [B, G, 3] directly (tail region of d_out).
// ---------------------------------------------------------------------------
__global__ __launch_bounds__(1024)
void fps_kernel(const float* __restrict__ xyz, float* __restrict__ center_out) {
  extern __shared__ float smem[];
  float* s_pts  = smem;                          // NPTS*3
  float* s_redd = smem + NPTS * 3;               // 32
  int*   s_redi = (int*)(s_redd + 32);           // 32

  const int b = blockIdx.x;
  const int t = threadIdx.x;
  const float* pb = xyz + (size_t)b * NPTS * 3;

  for (int i = t; i < NPTS * 3; i += 1024) s_pts[i] = pb[i];
  __syncthreads();

  const int PPT = NPTS / 1024;  // 8 points per thread (strided)
  float md[NPTS / 1024];
  #pragma unroll
  for (int j = 0; j < PPT; ++j) md[j] = 1e10f;

  float lx = s_pts[0], ly = s_pts[1], lz = s_pts[2];

  for (int g = 0; g < NUM_GROUP; ++g) {
    if (t == 0) {
      float* co = center_out + ((size_t)b * NUM_GROUP + g) * 3;
      co[0] = lx; co[1] = ly; co[2] = lz;
    }
    // update min-distances, track per-thread argmax (first-max tie semantics)
    float bd = -1.0f; int bi = 0;
    #pragma unroll
    for (int j = 0; j < PPT; ++j) {
      int n = t + j * 1024;
      float dx = s_pts[n * 3 + 0] - lx;
      float dy = s_pts[n * 3 + 1] - ly;
      float dz = s_pts[n * 3 + 2] - lz;
      float d  = dx * dx + dy * dy + dz * dz;
      float m  = md[j] < d ? md[j] : d;
      md[j] = m;
      if (m > bd) { bd = m; bi = n; }
    }
    wave_argmax(bd, bi);
    int wid = t >> 5;
    if ((t & 31) == 0) { s_redd[wid] = bd; s_redi[wid] = bi; }
    __syncthreads();
    if (t < 32) {
      bd = s_redd[t]; bi = s_redi[t];
      wave_argmax(bd, bi);
      if (t == 0) s_redi[0] = bi;
    }
    __syncthreads();
    int last = s_redi[0];
    lx = s_pts[last * 3 + 0];
    ly = s_pts[last * 3 + 1];
    lz = s_pts[last * 3 + 2];
    __syncthreads();  // protect s_redi[0] until everyone consumed it
  }
}

// ---------------------------------------------------------------------------
// Kernel 2: KNN (WMMA f32 16x16x4 score tiles) + top-32 + gather/recenter.
// Score(m,n) = |x_n|^2 - 2 c_m . x_n  (|c|^2 dropped: constant per row under
// top-k). A row m = (-2cx, -2cy, -2cz, 1), B col n = (x, y, z, |x|^2).
// One block = 16 centers; 8 waves, 4 WMMA tiles/wave/iter, each wave finally
// owns 2 centers for selection.
// ---------------------------------------------------------------------------
__global__ __launch_bounds__(256)
void knn_kernel(const float* __restrict__ xyz,
                const float* __restrict__ center,
                float* __restrict__ nb_out) {
  __shared__ float s_score[CTILE][SSTRIDE];
  __shared__ float s_c[CTILE][3];

  const int tile = blockIdx.x;              // 0 .. G/CTILE-1
  const int b    = blockIdx.y;              // 0 .. B-1
  const int t    = threadIdx.x;
  const int lane = t & 31;
  const int wave = t >> 5;
  const int g0   = tile * CTILE;

  const float* pb = xyz + (size_t)b * NPTS * 3;

  if (t < CTILE * 3)
    ((float*)s_c)[t] = center[((size_t)b * NUM_GROUP + g0) * 3 + t];
  __syncthreads();

  // A fragment (loop-invariant): 16x4 f32, lanes 0-15 hold K=0,1; 16-31 K=2,3
  const int m = lane & 15;
  v2f A;
  if (lane < 16) { A.x = -2.0f * s_c[m][0]; A.y = -2.0f * s_c[m][1]; }
  else           { A.x = -2.0f * s_c[m][2]; A.y = 1.0f; }

  // selection state: wave owns centers c0=2*wave, c1=2*wave+1
  float cd0 = 3.0e38f, cd1 = 3.0e38f;
  int   ci0 = 0,       ci1 = 0;
  float thr0 = 3.0e38f, thr1 = 3.0e38f;
  const int c0 = 2 * wave, c1 = 2 * wave + 1;

  const int mb = (lane >> 4) * 8;           // C-layout row base for this lane
  const int nc = lane & 15;                 // C-layout column for this lane

  for (int it = 0; it < NITER; ++it) {
    const int nbase = it * PTS_PER_ITER;

    #pragma unroll
    for (int tt = 0; tt < NT; ++tt) {
      // both half-waves read the same 16 points of this sub-tile
      const int n = nbase + wave * (16 * NT) + tt * 16 + (lane & 15);
      float px = pb[n * 3 + 0];
      float py = pb[n * 3 + 1];
      float pz = pb[n * 3 + 2];
      // speculative prefetch of next chunk (global_prefetch_b8); safe past end
      __builtin_prefetch(&pb[(n + PTS_PER_ITER) * 3], 0, 0);

      v2f B;
      if (lane < 16) { B.x = px; B.y = py; }
      else           { B.x = pz; B.y = px * px + py * py + pz * pz; }

      v8f C = {};
      C = __builtin_amdgcn_wmma_f32_16x16x4_f32(false, A, false, B,
                                                (short)0, C, false, false);

      // scatter C (16x16) to LDS: lane l vgpr v -> row v+8*(l>>4), col (l&15)
      const int colb = wave * (16 * NT) + tt * 16 + nc;
      #pragma unroll
      for (int v = 0; v < 8; ++v)
        s_score[mb + v][colb] = C[v];
    }
    __syncthreads();

    // selection: each wave scans the 512 scores of each of its 2 centers
    #pragma unroll 4
    for (int sub = 0; sub < PTS_PER_ITER / 32; ++sub) {
      const int col = sub * 32 + lane;
      topk_insert(s_score[c0][col], nbase + col, cd0, ci0, thr0, lane);
      topk_insert(s_score[c1][col], nbase + col, cd1, ci1, thr1, lane);
    }
    __syncthreads();
  }

  // sort candidates ascending (matches top_k nearest-first order), gather,
  // recenter, write: lane k -> neighborhood[b, g, k, :]
  bitonic32(cd0, ci0, lane);
  {
    const float* p = pb + (size_t)ci0 * 3;
    float* o = nb_out + (((size_t)(b * NUM_GROUP + g0 + c0)) * GROUP_SIZE + lane) * 3;
    o[0] = p[0] - s_c[c0][0];
    o[1] = p[1] - s_c[c0][1];
    o[2] = p[2] - s_c[c0][2];
  }
  bitonic32(cd1, ci1, lane);
  {
    const float* p = pb + (size_t)ci1 * 3;
    float* o = nb_out + (((size_t)(b * NUM_GROUP + g0 + c1)) * GROUP_SIZE + lane) * 3;
    o[0] = p[0] - s_c[c1][0];
    o[1] = p[1] - s_c[c1][1];
    o[2] = p[2] - s_c[c1][2];
  }
}

// ---------------------------------------------------------------------------
extern "C" void kernel_launch(void* const* d_in, const int* in_sizes, int n_in,
                              void* d_out, int out_size, void* d_ws, size_t ws_size,
                              hipStream_t stream) {
  (void)in_sizes; (void)n_in; (void)out_size; (void)d_ws; (void)ws_size;
  const float* xyz = (const float*)d_in[0];
  float* out    = (float*)d_out;
  float* nb     = out;                                               // [B,G,K,3]
  float* center = out + (size_t)BATCH * NUM_GROUP * GROUP_SIZE * 3;  // [B,G,3]

  size_t lds = (size_t)(NPTS * 3 + 32 + 32) * sizeof(float);
  fps_kernel<<<dim3(BATCH), dim3(1024), lds, stream>>>(xyz, center);

  dim3 grid(NUM_GROUP / CTILE, BATCH);
  knn_kernel<<<grid, dim3(256), 0, stream>>>(xyz, center, nb);
}